// ChamferDistance_55336358642037
// MI455X (gfx1250) — compile-verified
//
#include <hip/hip_runtime.h>
#include <hip/hip_bf16.h>

typedef __attribute__((ext_vector_type(2))) float v2f;
typedef __attribute__((ext_vector_type(8))) float v8f;

#define WAVES_PER_BLOCK 8
#define ROW_TILES 2  // 16x16 A-tiles per wave (32 source rows per wave)

// Raw v_min_num_f32: skips the sNaN canonicalization (v_max_num x,x) that
// fminf/llvm.minnum insert. All inputs here are finite by construction.
__device__ __forceinline__ float fmin_raw(float a, float b) {
  float r;
  asm("v_min_num_f32 %0, %1, %2" : "=v"(r) : "v"(a), "v"(b));
  return r;
}

// One wave owns 32 rows of `rows` and sweeps ALL of `cols`, producing the
// per-row min squared distance via V_WMMA_F32_16X16X4_F32:
//   A row n  = (r0, r1, r2, 1)            (K = 4, loop-invariant)
//   B col m  = (-2c0, -2c1, -2c2, |c|^2)  (rebuilt per column tile)
//   C[n][m]  = |r_n|^2                    (per-row, loop-invariant SRC2)
// D = A*B + C gives the exact 16x16 tile of squared distances.
__global__ __launch_bounds__(WAVES_PER_BLOCK * 32)
void chamfer_rowmin_wmma(const float* __restrict__ rows,  // [B][NR][3]
                         const float* __restrict__ cols,  // [B][NC][3]
                         float* __restrict__ row_min,     // [B][NR]
                         int B, int NR, int NC) {
  const int lane   = threadIdx.x & 31;
  const int waveId = threadIdx.x >> 5;
  const int gw     = blockIdx.x * WAVES_PER_BLOCK + waveId;
  const int strips = NR >> 5;    // 32-row strips (ROW_TILES == 2)
  if (gw >= B * strips) return;  // uniform per wave: EXEC all-ones inside

  const int b    = gw / strips;
  const int st   = gw - b * strips;
  const int base = st * (16 * ROW_TILES);

  const float* __restrict__ rbase = rows + (size_t)b * NR * 3;
  const float* __restrict__ cbase = cols + (size_t)b * NC * 3;

  const int  sub = lane & 15;
  const bool hi  = (lane >= 16);

  // ---- loop-invariant A tiles and per-row |r|^2 C vectors ----------------
  // f32 A 16x4 layout: lanes 0-15: VGPR0=K0, VGPR1=K1; lanes 16-31: K2, K3.
  // f32 C/D layout: VGPR i -> row i (lanes 0-15) / row 8+i (lanes 16-31).
  v2f a0, a1;
  v8f crow0, crow1;
  v8f acc0, acc1;
  {
    const int r0i = base + sub;
    const int r1i = base + 16 + sub;
    const float x0 = rbase[3 * r0i + 0], y0 = rbase[3 * r0i + 1], z0 = rbase[3 * r0i + 2];
    const float x1 = rbase[3 * r1i + 0], y1 = rbase[3 * r1i + 1], z1 = rbase[3 * r1i + 2];
    const float q0 = x0 * x0 + y0 * y0 + z0 * z0;
    const float q1 = x1 * x1 + y1 * y1 + z1 * z1;
    a0.x = hi ? z0 : x0;    // K2 : K0
    a0.y = hi ? 1.0f : y0;  // K3 : K1
    a1.x = hi ? z1 : x1;
    a1.y = hi ? 1.0f : y1;
#pragma unroll
    for (int i = 0; i < 8; ++i) {
      const int srcl = hi ? (8 + i) : i;  // both halves hold rows 0..15
      crow0[i] = __shfl(q0, srcl, 32);
      crow1[i] = __shfl(q1, srcl, 32);
      acc0[i]  = 3.0e38f;
      acc1[i]  = 3.0e38f;
    }
  }

  // One column-tile step: build B, issue BOTH WMMAs back-to-back (distinct
  // live D values so they overlap in the matrix pipe), then all 16 mins.
  auto tile_step = [&](float t0, float t1, float t2) {
    const float tsq = t0 * t0 + t1 * t1 + t2 * t2;
    // f32 B 4x16 layout (mirror of A): lanes 0-15: K0,K1; lanes 16-31: K2,K3.
    v2f bm;
    bm.x = hi ? (-2.0f * t2) : (-2.0f * t0);  // K2 : K0
    bm.y = hi ? tsq          : (-2.0f * t1);  // K3 : K1
    v8f d0 = __builtin_amdgcn_wmma_f32_16x16x4_f32(
        false, a0, false, bm, (short)0, crow0, false, false);
    v8f d1 = __builtin_amdgcn_wmma_f32_16x16x4_f32(
        false, a1, false, bm, (short)0, crow1, false, false);
#pragma unroll
    for (int i = 0; i < 8; ++i) acc0[i] = fmin_raw(acc0[i], d0[i]);
#pragma unroll
    for (int i = 0; i < 8; ++i) acc1[i] = fmin_raw(acc1[i], d1[i]);
  };

  // ---- sweep all column tiles, software-pipelined by one tile ------------
  const float* __restrict__ cp = cbase + 3 * (size_t)sub;  // my column's data
  float t0 = cp[0], t1 = cp[1], t2 = cp[2];                // preload tile 0
  for (int mt = 0; mt < NC - 16; mt += 16) {
    const float* __restrict__ np = cp + 48;  // next tile (always in-bounds)
    const float n0 = np[0];
    const float n1 = np[1];
    const float n2 = np[2];
    // Speculative prefetch ~3 tiles ahead (dropped silently if OOB).
    __builtin_prefetch((const void*)(np + 96), 0, 3);
    tile_step(t0, t1, t2);
    t0 = n0; t1 = n1; t2 = n2;
    cp = np;
  }
  tile_step(t0, t1, t2);  // peeled last tile

  // ---- per-row min: reduce across the 16 lanes of each half --------------
  // xor masks 1,2,4,8 never touch bit 4 -> halves reduce independently.
  v8f accs[2] = {acc0, acc1};
#pragma unroll
  for (int j = 0; j < ROW_TILES; ++j) {
#pragma unroll
    for (int i = 0; i < 8; ++i) {
      float v = accs[j][i];
      v = fmin_raw(v, __shfl_xor(v, 1, 32));
      v = fmin_raw(v, __shfl_xor(v, 2, 32));
      v = fmin_raw(v, __shfl_xor(v, 4, 32));
      v = fmin_raw(v, __shfl_xor(v, 8, 32));
      accs[j][i] = v;
    }
    float* __restrict__ out = row_min + (size_t)b * NR + base + 16 * j;
#pragma unroll
    for (int i = 0; i < 8; ++i) {
      if (lane == 0)  out[i]     = accs[j][i];  // rows 0..7 of tile j
      if (lane == 16) out[8 + i] = accs[j][i];  // rows 8..15 of tile j
    }
  }
}

// Single-block reduction: means of the two min arrays -> d_out[0], d_out[1].
__global__ __launch_bounds__(512)
void chamfer_reduce(const float* __restrict__ src_min,
                    const float* __restrict__ dst_min,
                    float* __restrict__ out, int n_src, int n_dst) {
  __shared__ float s0[512];
  __shared__ float s1[512];
  float a0 = 0.0f, a1 = 0.0f;
  for (int i = threadIdx.x; i < n_src; i += 512) a0 += src_min[i];
  for (int i = threadIdx.x; i < n_dst; i += 512) a1 += dst_min[i];
  s0[threadIdx.x] = a0;
  s1[threadIdx.x] = a1;
  __syncthreads();
  for (int off = 256; off > 0; off >>= 1) {
    if (threadIdx.x < off) {
      s0[threadIdx.x] += s0[threadIdx.x + off];
      s1[threadIdx.x] += s1[threadIdx.x + off];
    }
    __syncthreads();
  }
  if (threadIdx.x == 0) {
    out[0] = s0[0] / (float)n_src;
    out[1] = s1[0] / (float)n_dst;
  }
}

extern "C" void kernel_launch(void* const* d_in, const int* in_sizes, int n_in,
                              void* d_out, int out_size, void* d_ws, size_t ws_size,
                              hipStream_t stream) {
  const float* source = (const float*)d_in[0];  // [B][N][3] fp32
  const float* target = (const float*)d_in[1];  // [B][M][3] fp32

  const int B = 8;
  const int C = 3;
  const int N = in_sizes[0] / (B * C);  // 4096
  const int M = in_sizes[1] / (B * C);  // 4096

  float* src_min = (float*)d_ws;             // [B][N]
  float* dst_min = src_min + (size_t)B * N;  // [B][M]

  const int waves1  = B * (N >> 5);
  const int waves2  = B * (M >> 5);
  const int blocks1 = (waves1 + WAVES_PER_BLOCK - 1) / WAVES_PER_BLOCK;
  const int blocks2 = (waves2 + WAVES_PER_BLOCK - 1) / WAVES_PER_BLOCK;

  // Pass 1: src2dst — min over targets for each source row.
  chamfer_rowmin_wmma<<<blocks1, WAVES_PER_BLOCK * 32, 0, stream>>>(
      source, target, src_min, B, N, M);
  // Pass 2: dst2src — same kernel, roles swapped (d is symmetric).
  chamfer_rowmin_wmma<<<blocks2, WAVES_PER_BLOCK * 32, 0, stream>>>(
      target, source, dst_min, B, M, N);

  chamfer_reduce<<<1, 512, 0, stream>>>(src_min, dst_min, (float*)d_out,
                                        B * N, B * M);
}